// PositivityConstraint_89086211654295
// MI455X (gfx1250) — compile-verified
//
#include <hip/hip_runtime.h>
#include <math.h>
#include <stdint.h>

// ---------------------------------------------------------------------------
// G2 positivity constraint, MI455X (gfx1250) final kernel.
//
//   Bm_{ij} = (1/144) phi_{ikl} eps^{klmnpqr} phi_{jmn} phi_{pqr}
//   g = sym(Bm) / (|det Bm| + 1e-12)^(1/9);  out = sum relu(1e-6 - eig(g))
//
// Design: 32768 independent 7x7 problems -> one wave32 lane per element,
// everything register-resident. eps-contraction structure folded into
// constexpr sign/index tables (all lookups become immediates after unroll).
// Bulk input staging (256x35 floats / block) via one Tensor Data Mover
// descriptor per block (TENSORcnt path); per-lane async-LDS B128 fallback.
// WMMA intentionally unused: all matrix factors are per-element (<=7x21),
// and the dominant Jacobi eigensolve has no matmul shape.
// ---------------------------------------------------------------------------
struct G2Tab {
    signed char tri[7][7][7];   // sorted-triple index (lexicographic = np.combinations)
    signed char tsg[7][7][7];   // permutation sign, 0 if repeated index
    signed char ct[2401];       // [k][l][m][n] -> complement-triple index
    signed char cs[2401];       // sign of (k,l,m,n,comp) permutation, 0 if invalid
};

constexpr int triIndex(int a, int b, int c) {  // requires a<b<c
    int idx = 0;
    for (int x = 0; x < 7; ++x)
        for (int y = x + 1; y < 7; ++y)
            for (int z = y + 1; z < 7; ++z) {
                if (x == a && y == b && z == c) return idx;
                ++idx;
            }
    return -1;
}

constexpr G2Tab makeTab() {
    G2Tab t{};
    for (int i = 0; i < 7; ++i)
        for (int j = 0; j < 7; ++j)
            for (int k = 0; k < 7; ++k) {
                if (i == j || j == k || i == k) { t.tri[i][j][k] = 0; t.tsg[i][j][k] = 0; continue; }
                int a = i, b = j, c = k, s = 1, tmp = 0;
                if (a > b) { tmp = a; a = b; b = tmp; s = -s; }
                if (b > c) { tmp = b; b = c; c = tmp; s = -s; }
                if (a > b) { tmp = a; a = b; b = tmp; s = -s; }
                t.tri[i][j][k] = (signed char)triIndex(a, b, c);
                t.tsg[i][j][k] = (signed char)s;
            }
    for (int k = 0; k < 7; ++k)
        for (int l = 0; l < 7; ++l)
            for (int m = 0; m < 7; ++m)
                for (int n = 0; n < 7; ++n) {
                    int id = ((k * 7 + l) * 7 + m) * 7 + n;
                    if (k == l || k == m || k == n || l == m || l == n || m == n) {
                        t.ct[id] = 0; t.cs[id] = 0; continue;
                    }
                    int comp[3] = {0, 0, 0}; int ci = 0;
                    for (int x = 0; x < 7; ++x)
                        if (x != k && x != l && x != m && x != n) comp[ci++] = x;
                    int seq[7] = {k, l, m, n, comp[0], comp[1], comp[2]};
                    int inv = 0;
                    for (int x = 0; x < 7; ++x)
                        for (int y = x + 1; y < 7; ++y)
                            if (seq[x] > seq[y]) ++inv;
                    t.ct[id] = (signed char)triIndex(comp[0], comp[1], comp[2]);
                    t.cs[id] = (signed char)((inv & 1) ? -1 : 1);
                }
    return t;
}

constexpr G2Tab TAB = makeTab();

#define BLOCK 256
#define ROW   35                     // floats per element
#define NDW   (BLOCK * ROW)          // dwords per block tile = 8960
#define SWEEPS 8

typedef unsigned int u32x4 __attribute__((ext_vector_type(4)));
typedef int          i32x4 __attribute__((ext_vector_type(4)));
typedef int          i32x8 __attribute__((ext_vector_type(8)));

__global__ __launch_bounds__(BLOCK) void g2_positivity_kernel(
    const float* __restrict__ phi, float* __restrict__ out, int n)
{
    __shared__ __align__(16) float lphi[NDW];   // 35 KB of the 320 KB/WGP LDS
    const unsigned t   = threadIdx.x;
    const int      gid = (int)(blockIdx.x * BLOCK + t);
    const bool fullBlock = ((int)((blockIdx.x + 1) * BLOCK) <= n);

    float ph[ROW];

    if (fullBlock) {
        const unsigned long long ga =
            (unsigned long long)(uintptr_t)(phi + (size_t)blockIdx.x * NDW);
        const unsigned lbase = (unsigned)(uintptr_t)&lphi[0];   // LDS aperture low 32 bits

#if __has_builtin(__builtin_amdgcn_tensor_load_to_lds)
        // ---- Tensor Data Mover: one D# moves the whole 35840B tile (2D: 8960x1 dwords)
        if (t == 0) {
            u32x4 g0;
            g0.x = 1u;                                             // count=1, no gather/iterate
            g0.y = lbase;                                          // lds_addr [63:32]
            g0.z = (unsigned)(ga & 0xFFFFFFFFull);                 // global_addr [95:64]
            g0.w = (unsigned)((ga >> 32) & 0x01FFFFFFull)          // global_addr [120:96]
                 | 0x80000000u;                                    // type=2 ("image") [127:126]
            i32x8 g1;
            g1[0] = (int)(2u << 16);                               // data_size=4B; mask=0; no pad
            g1[1] = (int)((NDW & 0xFFFFu) << 16);                  // tensor_dim0 low16 @ [63:48]
            g1[2] = (int)(((NDW >> 16) & 0xFFFFu) | (1u << 16));   // dim0 hi16 | tensor_dim1 lo16
            g1[3] = (int)((NDW & 0xFFFFu) << 16);                  // tensor_dim1 hi16=0 | tile_dim0
            g1[4] = 1;                                             // tile_dim1=1, tile_dim2=0
            g1[5] = (int)NDW;                                      // tensor_dim0_stride low32
            g1[6] = (int)((NDW & 0xFFFFu) << 16);                  // stride0 hi16=0 | stride1 lo16
            g1[7] = (int)(NDW >> 16);                              // stride1 hi32
            i32x4 g2 = {0, 0, 0, 0};                               // dims 2/3 unused
            i32x4 g3 = {0, 0, 0, 0};                               // dim 4 unused
#if __clang_major__ >= 23
            i32x8 g1b = {0, 0, 0, 0, 0, 0, 0, 0};
            __builtin_amdgcn_tensor_load_to_lds(g0, g1, g2, g3, g1b, 0);
#else
            __builtin_amdgcn_tensor_load_to_lds(g0, g1, g2, g3, 0);
#endif
            __builtin_amdgcn_s_wait_tensorcnt(0);
        }
        __syncthreads();
#else
        // ---- fallback: per-lane async global->LDS B128 copy (ASYNCcnt path) ----
        #pragma unroll
        for (int k = 0; k < 8; ++k) {
            const unsigned idx = t + 256u * (unsigned)k;           // 16B chunk index
            const unsigned long long gac = ga + (unsigned long long)idx * 16ull;
            const unsigned la = lbase + idx * 16u;
            asm volatile("global_load_async_to_lds_b128 %0, %1, off"
                         :: "v"(la), "v"(gac) : "memory");
        }
        if (t < (unsigned)(NDW / 4 - 2048)) {                      // tail chunks
            const unsigned idx = 2048u + t;
            const unsigned long long gac = ga + (unsigned long long)idx * 16ull;
            const unsigned la = lbase + idx * 16u;
            asm volatile("global_load_async_to_lds_b128 %0, %1, off"
                         :: "v"(la), "v"(gac) : "memory");
        }
        asm volatile("s_wait_asynccnt 0x0" ::: "memory");
        __syncthreads();
#endif
        #pragma unroll
        for (int c = 0; c < ROW; ++c) ph[c] = lphi[t * ROW + c];   // stride 35 (odd) -> conflict-free
    } else {
        if (gid < n) {
            #pragma unroll
            for (int c = 0; c < ROW; ++c) ph[c] = phi[(size_t)gid * ROW + c];
        }
    }
    if (gid >= n) return;

    // ---- cubic contraction, fully unrolled; M = 6*Bm (scale hoisted to the end)
    float M[7][7];
    #pragma unroll
    for (int j = 0; j < 7; ++j) {
        float Tq[21];
        {
            int kl = 0;
            #pragma unroll
            for (int k = 0; k < 7; ++k) {
                #pragma unroll
                for (int l = k + 1; l < 7; ++l) {
                    float s = 0.0f;
                    #pragma unroll
                    for (int m = 0; m < 7; ++m) {
                        #pragma unroll
                        for (int nn = m + 1; nn < 7; ++nn) {
                            const int id  = ((k * 7 + l) * 7 + m) * 7 + nn;
                            const int c_s = TAB.cs[id];
                            const int p_s = TAB.tsg[j][m][nn];
                            if (c_s != 0 && p_s != 0) {
                                s += (float)(c_s * p_s) *
                                     ph[TAB.ct[id]] * ph[TAB.tri[j][m][nn]];
                            }
                        }
                    }
                    Tq[kl++] = s;
                }
            }
        }
        #pragma unroll
        for (int i = 0; i < 7; ++i) {
            float acc = 0.0f;
            int kl = 0;
            #pragma unroll
            for (int k = 0; k < 7; ++k) {
                #pragma unroll
                for (int l = k + 1; l < 7; ++l) {
                    const int g_s = TAB.tsg[i][k][l];
                    if (g_s != 0)
                        acc += (float)g_s * ph[TAB.tri[i][k][l]] * Tq[kl];
                    ++kl;
                }
            }
            M[i][j] = acc;                    // = 6 * Bm_{ij}
        }
    }

    // ---- symmetrize in place (matches 0.5*(g + g^T))
    #pragma unroll
    for (int i = 0; i < 7; ++i)
        #pragma unroll
        for (int j = i + 1; j < 7; ++j) {
            const float s = 0.5f * (M[i][j] + M[j][i]);
            M[i][j] = s; M[j][i] = s;
        }

    // ---- cyclic Jacobi eigensolve (branchless safe rotations), sweep loop kept rolled
    #pragma unroll 1
    for (int sweep = 0; sweep < SWEEPS; ++sweep) {
        #pragma unroll
        for (int p = 0; p < 7; ++p) {
            #pragma unroll
            for (int q = p + 1; q < 7; ++q) {
                const float apq = M[p][q];
                const float app = M[p][p];
                const float aqq = M[q][q];
                const float absapq = fabsf(apq);
                const float theta  = (aqq - app) / (2.0f * apq);   // may be inf/NaN
                const float at     = fabsf(theta);
                const float tmag   = 1.0f / (at + sqrtf(1.0f + at * at));
                float tt = copysignf(tmag, theta);
                tt = (absapq > 1e-30f) ? tt : 0.0f;                // kills 0/0 NaN
                const float cc = rsqrtf(1.0f + tt * tt);
                const float ss = tt * cc;
                #pragma unroll
                for (int r = 0; r < 7; ++r) {
                    if (r != p && r != q) {
                        const float arp = M[r][p];
                        const float arq = M[r][q];
                        const float nrp = cc * arp - ss * arq;
                        const float nrq = ss * arp + cc * arq;
                        M[r][p] = nrp; M[p][r] = nrp;
                        M[r][q] = nrq; M[q][r] = nrq;
                    }
                }
                const float tapq = tt * apq;
                M[p][p] = app - tapq;
                M[q][q] = aqq + tapq;
                M[p][q] = 0.0f; M[q][p] = 0.0f;
            }
        }
    }

    // ---- det / normalization; undo the hoisted factor 6:
    //      lam_true = lam/6, det_true = det/6^7
    float det = 1.0f;
    #pragma unroll
    for (int i = 0; i < 7; ++i) det *= M[i][i];
    const float detTrue = fabsf(det) * (1.0f / 279936.0f);         // 6^7
    const float inv = (1.0f / 6.0f) / powf(detTrue + 1e-12f, 1.0f / 9.0f);

    float res = 0.0f;
    #pragma unroll
    for (int i = 0; i < 7; ++i) {
        const float v = 1e-6f - M[i][i] * inv;
        res += (v > 0.0f) ? v : 0.0f;
    }
    out[gid] = res;
}

extern "C" void kernel_launch(void* const* d_in, const int* in_sizes, int n_in,
                              void* d_out, int out_size, void* d_ws, size_t ws_size,
                              hipStream_t stream)
{
    (void)n_in; (void)d_ws; (void)ws_size; (void)out_size;
    const float* phi = (const float*)d_in[0];
    float* out = (float*)d_out;
    const int n = in_sizes[0] / ROW;               // batch = 32768
    const int grid = (n + BLOCK - 1) / BLOCK;      // 128 blocks of 256 (8 wave32)
    g2_positivity_kernel<<<grid, BLOCK, 0, stream>>>(phi, out, n);
}